// Triplet_Loss_31516470018601
// MI455X (gfx1250) — compile-verified
//
#include <hip/hip_runtime.h>
#include <math.h>

#define L_P   8192
#define DIMD  512
#define EPSF  1e-6f
#define MARGINF 0.05f

// 1/sqrt(512)
#define SCALEF 0.044194173824159216f

typedef float v2f __attribute__((ext_vector_type(2)));
typedef float v8f __attribute__((ext_vector_type(8)));

// ---------------------------------------------------------------------------
// Kernel 0: zero the scalar output
// ---------------------------------------------------------------------------
__global__ void tl_zero(float* out) {
    if (threadIdx.x == 0) out[0] = 0.0f;
}

// ---------------------------------------------------------------------------
// Kernel 1: per-row constants.
//   ra[i] = ||x_i||^2 + 2*eps*sum(x_i)          (row term of dsq)
//   ca[j] = ||x_j||^2 - 2*eps*sum(x_j)          (col term of dsq)
//   rb[i] = MARGIN - d_pn[i]                     (row term of the relu)
// One wave (32 lanes) per row; lane reads 4x float4.
// ---------------------------------------------------------------------------
__global__ __launch_bounds__(256) void tl_row_stats(
    const float* __restrict__ xp, const float* __restrict__ xn,
    const int* __restrict__ tptr,
    float* __restrict__ ra, float* __restrict__ ca, float* __restrict__ rb)
{
    const int wid  = threadIdx.x >> 5;
    const int lane = threadIdx.x & 31;
    const int row  = blockIdx.x * 8 + wid;
    if (row >= L_P) return;

    const float* xr  = xp + (size_t)row * DIMD;
    const float* xnr = xn + (size_t)tptr[0] * DIMD;

    float ssq = 0.0f, ss = 0.0f, dn = 0.0f;
    #pragma unroll
    for (int w = 0; w < 4; ++w) {
        const int c = lane * 4 + w * 128;
        float4 x = *(const float4*)(xr + c);
        float4 n = *(const float4*)(xnr + c);
        ssq += x.x * x.x + x.y * x.y + x.z * x.z + x.w * x.w;
        ss  += x.x + x.y + x.z + x.w;
        float dx = x.x - n.x + EPSF;
        float dy = x.y - n.y + EPSF;
        float dz = x.z - n.z + EPSF;
        float dw = x.w - n.w + EPSF;
        dn += dx * dx + dy * dy + dz * dz + dw * dw;
    }
    #pragma unroll
    for (int m = 16; m >= 1; m >>= 1) {
        ssq += __shfl_xor(ssq, m, 32);
        ss  += __shfl_xor(ss,  m, 32);
        dn  += __shfl_xor(dn,  m, 32);
    }
    if (lane == 0) {
        ra[row] = ssq + 2.0f * EPSF * ss;
        ca[row] = ssq - 2.0f * EPSF * ss;
        rb[row] = MARGINF - sqrtf(dn) * SCALEF;
    }
}

// ---------------------------------------------------------------------------
// Kernel 2: fused gram-GEMM + triplet epilogue.
// Block = 256 threads (8 waves), block tile 128x128, BK=32 K-chunks staged in
// double-buffered LDS via GLOBAL_LOAD_ASYNC_TO_LDS_B128 (ASYNCcnt-tracked DMA,
// no VGPR round-trip). Prefetch of chunk c+1 overlaps the 64 WMMAs of chunk c.
// Wave tile 64x32 = 4x2 tiles of V_WMMA_F32_16X16X4_F32.
// ---------------------------------------------------------------------------
#define BM 128
#define BN 128
#define BK 32
#define LSTR 36                 // padded LDS row stride (floats), 144B = 9*16B
#define BUFELTS (BM * LSTR)

__global__ __launch_bounds__(256) void tl_gram_loss(
    const float* __restrict__ xp,
    const float* __restrict__ ra, const float* __restrict__ ca,
    const float* __restrict__ rb, float* __restrict__ out)
{
    __shared__ float As[2][BUFELTS];
    __shared__ float Bs[2][BUFELTS];
    __shared__ float wsum[8];

    const int gi0 = blockIdx.y * BM;
    const int gj0 = blockIdx.x * BN;
    const int tid  = threadIdx.x;
    const int wid  = tid >> 5;
    const int lane = tid & 31;
    const int wm = wid & 1;   // 0..1  -> 64-row slice of M
    const int wn = wid >> 1;  // 0..3  -> 32-col slice of N

    // Per-thread staging addresses: 4 b128 slots per operand per chunk.
    // LDS generic pointers truncate to the wave-relative LDS byte offset.
    unsigned ldsA[4], ldsB[4];
    unsigned long long gA[4], gB[4];
    #pragma unroll
    for (int it = 0; it < 4; ++it) {
        const int q = tid + it * 256;       // 0..1023 float4 slots
        const int r = q >> 3;
        const int c = (q & 7) * 4;
        ldsA[it] = (unsigned)(uintptr_t)(&As[0][r * LSTR + c]);
        ldsB[it] = (unsigned)(uintptr_t)(&Bs[0][r * LSTR + c]);
        gA[it] = (unsigned long long)(uintptr_t)(xp + (size_t)(gi0 + r) * DIMD + c);
        gB[it] = (unsigned long long)(uintptr_t)(xp + (size_t)(gj0 + r) * DIMD + c);
    }
    const unsigned bufbytes = (unsigned)(BUFELTS * sizeof(float));

    auto stage = [&](int k0, int buf) {
        const unsigned long long koff = (unsigned long long)k0 * 4ull;
        const unsigned boff = buf ? bufbytes : 0u;
        #pragma unroll
        for (int it = 0; it < 4; ++it) {
            asm volatile("global_load_async_to_lds_b128 %0, %1, off"
                         :: "v"(ldsA[it] + boff), "v"(gA[it] + koff) : "memory");
            asm volatile("global_load_async_to_lds_b128 %0, %1, off"
                         :: "v"(ldsB[it] + boff), "v"(gB[it] + koff) : "memory");
        }
    };

    v8f acc[4][2];
    #pragma unroll
    for (int mt = 0; mt < 4; ++mt)
        #pragma unroll
        for (int nt = 0; nt < 2; ++nt)
            #pragma unroll
            for (int v = 0; v < 8; ++v) acc[mt][nt][v] = 0.0f;

    const int frow = lane & 15;
    const int fk   = (lane >> 4) * 2;

    stage(0, 0);
    asm volatile("s_wait_asynccnt 0x0" ::: "memory");
    __syncthreads();

    const int NCH = DIMD / BK;   // 16 chunks
    for (int ch = 0; ch < NCH; ++ch) {
        const int cur = ch & 1;
        if (ch + 1 < NCH) stage((ch + 1) * BK, cur ^ 1);   // async prefetch

        const float* Ab = As[cur];
        const float* Bb = Bs[cur];
        #pragma unroll
        for (int kk = 0; kk < BK; kk += 4) {
            v2f a[4], b[2];
            #pragma unroll
            for (int mt = 0; mt < 4; ++mt)
                a[mt] = *(const v2f*)(Ab + (wm * 64 + mt * 16 + frow) * LSTR + kk + fk);
            #pragma unroll
            for (int nt = 0; nt < 2; ++nt)
                b[nt] = *(const v2f*)(Bb + (wn * 32 + nt * 16 + frow) * LSTR + kk + fk);
            #pragma unroll
            for (int mt = 0; mt < 4; ++mt)
                #pragma unroll
                for (int nt = 0; nt < 2; ++nt)
                    acc[mt][nt] = __builtin_amdgcn_wmma_f32_16x16x4_f32(
                        false, a[mt], false, b[nt],
                        (short)0, acc[mt][nt], false, false);
        }

        asm volatile("s_wait_asynccnt 0x0" ::: "memory");
        __syncthreads();
    }

    // fused epilogue: dsq -> d_pp -> relu -> partial sum
    const float c0 = (float)DIMD * EPSF * EPSF;
    const int mhalf = (lane >> 4) * 8;     // C layout: lanes 16-31 hold M=v+8
    const int nloc  = lane & 15;
    float part = 0.0f;
    #pragma unroll
    for (int mt = 0; mt < 4; ++mt) {
        #pragma unroll
        for (int nt = 0; nt < 2; ++nt) {
            const int gj = gj0 + wn * 32 + nt * 16 + nloc;
            const float cav = ca[gj];
            #pragma unroll
            for (int v = 0; v < 8; ++v) {
                const int gi = gi0 + wm * 64 + mt * 16 + v + mhalf;
                const float g = acc[mt][nt][v];
                const float dsq = ra[gi] + cav - 2.0f * g + c0;
                const float dpp = sqrtf(fmaxf(dsq, 1e-12f)) * SCALEF;
                part += fmaxf(0.0f, dpp + rb[gi]);
            }
        }
    }
    #pragma unroll
    for (int m = 16; m >= 1; m >>= 1) part += __shfl_xor(part, m, 32);
    if (lane == 0) wsum[wid] = part;
    __syncthreads();
    if (tid == 0) {
        float tot = 0.0f;
        #pragma unroll
        for (int w = 0; w < 8; ++w) tot += wsum[w];
        unsafeAtomicAdd(out, tot * (1.0f / (float)(L_P - 1)));
    }
}

// ---------------------------------------------------------------------------
extern "C" void kernel_launch(void* const* d_in, const int* in_sizes, int n_in,
                              void* d_out, int out_size, void* d_ws, size_t ws_size,
                              hipStream_t stream) {
    const float* xp = (const float*)d_in[0];   // [8192, 512] f32
    const float* xn = (const float*)d_in[1];   // [1024, 512] f32
    const int*   t  = (const int*)d_in[2];     // scalar index

    float* out = (float*)d_out;
    float* ra = (float*)d_ws;                  // 8192 f32
    float* ca = ra + L_P;                      // 8192 f32
    float* rb = ca + L_P;                      // 8192 f32

    tl_zero<<<1, 64, 0, stream>>>(out);
    tl_row_stats<<<L_P / 8, 256, 0, stream>>>(xp, xn, t, ra, ca, rb);
    dim3 grid(L_P / BN, L_P / BM);
    tl_gram_loss<<<grid, 256, 0, stream>>>(xp, ra, ca, rb, out);
}